// QuantumRegressionHybrid_65481071407160
// MI455X (gfx1250) — compile-verified
//
#include <hip/hip_runtime.h>

#define NW    14
#define DIM   16384          // 2^14 amplitudes per row
#define BATCH 512
#define SPAD  (DIM + (DIM >> 4) + (DIM >> 8) + 8)   // bank-conflict padding

typedef __attribute__((ext_vector_type(2))) float v2f;
typedef __attribute__((ext_vector_type(8))) float v8f;

// padding map: e -> e + e/16 + e/256.  For operands with disjoint bit fields
// it is additive: padc(a|b) = padc(a) + padc(b).
__device__ __host__ constexpr int padc(int e) { return e + (e >> 4) + (e >> 8); }

// Quadratic form  sum_fibers  x^H H x  for one group, fiber stride = 1<<S.
// col/khalf follow the V_WMMA_F32_16X16X4_F32 register layouts.
template<int S>
__device__ __forceinline__ float group_quadform(const float* __restrict__ sre,
                                                const float* __restrict__ sim,
                                                const float* __restrict__ hr,
                                                const float* __restrict__ hi,
                                                int col, int khalf, int tile0)
{
    // A fragments (16x4 per k-step): lane row M=col, K = ks*4 + khalf*2 + {0,1}
    v2f Ar[4], Ai[4];
#pragma unroll
    for (int ks = 0; ks < 4; ++ks) {
        const int kk = (col << 4) + ks * 4 + khalf * 2;
        Ar[ks].x = hr[kk]; Ar[ks].y = hr[kk + 1];
        Ai[ks].x = hi[kk]; Ai[ks].y = hi[kk + 1];
    }
    // loop-invariant half-select components (disjoint bit fields -> additive pad)
    const int cB = khalf ? padc(2 << S) : 0;   // K-half (bit S+1) for B loads
    const int cD = khalf ? padc(8 << S) : 0;   // M-half (bit S+3) for dot loads

    float acc = 0.f;
    for (int tile = tile0; tile < tile0 + 32; ++tile) {
        const int f  = (tile << 4) + col;                       // this lane's fiber
        const int cb = ((f >> S) << (S + 4)) | (f & ((1 << S) - 1));
        const int pb = cb + (cb >> 4) + (cb >> 8);              // padded fiber base
        const float* br0 = sre + pb + cB;
        const float* bi0 = sim + pb + cB;
        const float* dr0 = sre + pb + cD;
        const float* di0 = sim + pb + cD;

        v8f accU = {}, accV = {}, accW = {};
#pragma unroll
        for (int ks = 0; ks < 4; ++ks) {
            v2f br, bi;                       // compile-time ds immediate offsets
            br.x = br0[padc((ks * 4 + 0) << S)];
            br.y = br0[padc((ks * 4 + 1) << S)];
            bi.x = bi0[padc((ks * 4 + 0) << S)];
            bi.y = bi0[padc((ks * 4 + 1) << S)];
            // u = Hr*xr ; v = Hr*xi ; w = Hi*xr  (Hermitian: 3 GEMMs instead of 4)
            accU = __builtin_amdgcn_wmma_f32_16x16x4_f32(false, Ar[ks], false, br,
                                                         (short)0, accU, false, false);
            accV = __builtin_amdgcn_wmma_f32_16x16x4_f32(false, Ar[ks], false, bi,
                                                         (short)0, accV, false, false);
            accW = __builtin_amdgcn_wmma_f32_16x16x4_f32(false, Ai[ks], false, br,
                                                         (short)0, accW, false, false);
        }
        // <x|H|x> contribution: xr.u + xi.(v + 2w), C/D layout: M = r + khalf*8
#pragma unroll
        for (int r = 0; r < 8; ++r) {
            const float xr = dr0[padc(r << S)];
            const float xi = di0[padc(r << S)];
            acc += xr * accU[r] + xi * (accV[r] + 2.0f * accW[r]);
        }
    }
    return acc;
}

// One workgroup (8 wave32) per batch row; whole 128 KB row lives in LDS.
__global__ __launch_bounds__(256)
void qc_wmma_kernel(const float* __restrict__ g_re, const float* __restrict__ g_im,
                    const float* __restrict__ rx,   const float* __restrict__ ry,
                    const float* __restrict__ hw,   const float* __restrict__ hb,
                    float* __restrict__ out)
{
    __shared__ float s_re[SPAD];
    __shared__ float s_im[SPAD];
    __shared__ float s_Hr[4][256];           // group Hamiltonians, real part
    __shared__ float s_Hi[4][256];           // imag part (antisymmetric)
    __shared__ float s_wA[NW], s_wCr[NW], s_wCi[NW];
    __shared__ float s_red[8];

    const int tid = threadIdx.x;
    const int row = blockIdx.x;

    // ---- Heisenberg picture: M_w = U_w^dag Z U_w, folded with head weight h_w.
    if (tid < NW) {
        float cxa = cosf(rx[tid]), sxa = sinf(rx[tid]);
        float cya = cosf(ry[tid]), sya = sinf(ry[tid]);
        float h = hw[tid];
        s_wA[tid]  = h * (cxa * cya);          // diagonal: +a / -a
        s_wCr[tid] = h * (-sya);               // c = -sin(ry) - i sin(rx) cos(ry)
        s_wCi[tid] = h * (-sxa * cya);
    }
    __syncthreads();

    // ---- build 16x16 Hermitian H_g: groups {w0-3},{w4-7},{w8-11},{w12,13 (x I4)}
    for (int e = tid; e < 1024; e += 256) {
        const int g = e >> 8, t = (e >> 4) & 15, tp = e & 15;
        float hrv = 0.f, hiv = 0.f;
        const int nj = (g < 3) ? 4 : 2;
        for (int j = 0; j < nj; ++j) {
            const int w = (g < 3) ? (g * 4 + j) : (12 + j);
            const int p = (g < 3) ? (3 - j) : (1 - j);
            const int mask = 15 ^ (1 << p);
            if ((t & mask) == (tp & mask)) {
                const int tb = (t >> p) & 1, tpb = (tp >> p) & 1;
                if (tb == tpb)      hrv += tb ? -s_wA[w] : s_wA[w];
                else if (tb == 0) { hrv += s_wCr[w]; hiv += s_wCi[w]; }  // M[0][1]=c
                else              { hrv += s_wCr[w]; hiv -= s_wCi[w]; }  // M[1][0]=c*
            }
        }
        s_Hr[g][(t << 4) | tp] = hrv;
        s_Hi[g][(t << 4) | tp] = hiv;
    }

    // ---- CNOT chain == Gray-code permutation: state[i] = in[i ^ (i>>1)].
    //      b128 coalesced global reads, scatter to i = gray^{-1}(j) (prefix-xor).
    const float* __restrict__ pre = g_re + row * DIM;
    const float* __restrict__ pim = g_im + row * DIM;
    for (int base = tid * 4; base < DIM; base += 1024) {
        const float4 r4 = *(const float4*)(pre + base);
        const float4 i4 = *(const float4*)(pim + base);
#pragma unroll
        for (int q = 0; q < 4; ++q) {
            int i = base + q;
            i ^= i >> 1; i ^= i >> 2; i ^= i >> 4; i ^= i >> 8;
            const int pe = i + (i >> 4) + (i >> 8);
            s_re[pe] = (q == 0) ? r4.x : (q == 1) ? r4.y : (q == 2) ? r4.z : r4.w;
            s_im[pe] = (q == 0) ? i4.x : (q == 1) ? i4.y : (q == 2) ? i4.z : i4.w;
        }
    }
    __syncthreads();

    // ---- 2 waves per group, 32 tiles (of 16 fibers) each
    const int lane  = tid & 31;
    const int wave  = tid >> 5;
    const int col   = lane & 15;
    const int khalf = lane >> 4;
    const int tile0 = (wave & 1) * 32;
    // group id is wave-uniform: force scalar branches (EXEC stays all-1s for WMMA)
    const int gs = __builtin_amdgcn_readfirstlane(wave >> 1);

    float acc;
    if (gs == 0)      acc = group_quadform<10>(s_re, s_im, &s_Hr[0][0], &s_Hi[0][0], col, khalf, tile0);
    else if (gs == 1) acc = group_quadform<6>(s_re, s_im, &s_Hr[1][0], &s_Hi[1][0], col, khalf, tile0);
    else if (gs == 2) acc = group_quadform<2>(s_re, s_im, &s_Hr[2][0], &s_Hi[2][0], col, khalf, tile0);
    else              acc = group_quadform<0>(s_re, s_im, &s_Hr[3][0], &s_Hi[3][0], col, khalf, tile0);

    // ---- reduce: wave32 shuffle tree, then cross-wave via LDS
    for (int off = 16; off >= 1; off >>= 1)
        acc += __shfl_xor(acc, off, 32);
    if (lane == 0) s_red[wave] = acc;
    __syncthreads();
    if (tid == 0) {
        float tot = hb[0];
        for (int i = 0; i < 8; ++i) tot += s_red[i];
        out[row] = tot;
    }
}

extern "C" void kernel_launch(void* const* d_in, const int* in_sizes, int n_in,
                              void* d_out, int out_size, void* d_ws, size_t ws_size,
                              hipStream_t stream) {
    (void)in_sizes; (void)n_in; (void)d_ws; (void)ws_size; (void)out_size;
    const float* state_re = (const float*)d_in[0];
    const float* state_im = (const float*)d_in[1];
    const float* rx       = (const float*)d_in[2];
    const float* ry       = (const float*)d_in[3];
    const float* head_w   = (const float*)d_in[4];
    const float* head_b   = (const float*)d_in[5];
    float* out = (float*)d_out;
    qc_wmma_kernel<<<BATCH, 256, 0, stream>>>(state_re, state_im, rx, ry,
                                              head_w, head_b, out);
}